// DAGModel_13735305412941
// MI455X (gfx1250) — compile-verified
//
#include <hip/hip_runtime.h>

// MI455X / gfx1250: wave32, WMMA f32 16x16x4 (exact fp32 matrix path).
typedef float v2f __attribute__((ext_vector_type(2)));
typedef float v8f __attribute__((ext_vector_type(8)));

#define Bsz 64
#define Hd 128
#define Pn 1000
#define Dd 20
#define MPar 8
#define Nnodes 20001
#define NV_STRIDE (Bsz * Hd)   // 8192 floats per node row-block

// node state nv: (node, B, H) f32, nodes 0..20001 (0=zeros, 1=embedding)

__global__ __launch_bounds__(256) void dag_init_kernel(const float* __restrict__ emb,
                                                       float* __restrict__ nv) {
    int i = blockIdx.x * 256 + threadIdx.x;
    if (i < Bsz * Hd) {
        nv[i] = 0.0f;                  // node 0 = zero vector (parent padding)
        nv[NV_STRIDE + i] = emb[i];    // node 1 = input embedding, (B,H) row-major
    }
}

__global__ __launch_bounds__(256) void dag_depth_kernel(
    const int* __restrict__ node_indices,
    const int* __restrict__ parent_indices,
    const float* __restrict__ emb_table,
    const float* __restrict__ W1, const float* __restrict__ b1,
    const float* __restrict__ W2, const float* __restrict__ b2,
    float* __restrict__ nv, int d)
{
    __shared__ float pv[32][132];     // parent-sum tile (pad 4 -> conflict-free col reads)
    __shared__ float hbuf[32][132];   // relu(h1) tile
    __shared__ float ne_s[Hd];        // node embedding (broadcast over batch)
    __shared__ int   par_s[MPar];

    const int p    = blockIdx.x;      // node-within-depth 0..999
    const int by   = blockIdx.y;      // batch half 0..1 (rows by*32..by*32+31)
    const int t    = threadIdx.x;
    const int node = node_indices[d * Pn + p];   // == 2 + d*1000 + p

    if (t < MPar) par_s[t] = parent_indices[(d * Pn + p) * MPar + t];
    if (t < Hd)   ne_s[t]  = emb_table[(size_t)node * Hd + t];
    __syncthreads();

    int pr[MPar];
#pragma unroll
    for (int m = 0; m < MPar; ++m) pr[m] = par_s[m];

    // ---- gather: pv[b_local][h] = sum over 8 parents (index 0 row is zeros)
    const size_t bofs = (size_t)by * (32 * Hd);
    for (int i = t; i < 32 * Hd; i += 256) {
        float s = 0.f;
#pragma unroll
        for (int m = 0; m < MPar; ++m)
            s += nv[(size_t)pr[m] * NV_STRIDE + bofs + i];
        pv[i >> 7][i & 127] = s;
    }
    __syncthreads();

    const int lane = t & 31;
    const int w    = t >> 5;         // wave id: output column tile (8 x 16 = 128 cols)
    const int half = lane >> 4;      // 0|1
    const int l16  = lane & 15;
    const int kb   = half * 2;       // K sub-offset inside 4-wide chunk (ISA A/B layout)
    const int cc   = w * 16 + l16;   // output column

    // ---- GEMM1: h = relu([pv | ne] (32x256) @ W1^T (256x128) + b1)
    v8f acc0 = {};
    v8f acc1 = {};
    const float* w1p = W1 + (size_t)cc * 256 + kb;
    for (int k0 = 0; k0 < 128; k0 += 4) {            // K = 0..127 from pv
        v2f bf; bf.x = w1p[k0]; bf.y = w1p[k0 + 1];
        v2f a0; a0.x = pv[l16][k0 + kb];      a0.y = pv[l16][k0 + kb + 1];
        v2f a1; a1.x = pv[16 + l16][k0 + kb]; a1.y = pv[16 + l16][k0 + kb + 1];
        acc0 = __builtin_amdgcn_wmma_f32_16x16x4_f32(false, a0, false, bf, (short)0, acc0, false, false);
        acc1 = __builtin_amdgcn_wmma_f32_16x16x4_f32(false, a1, false, bf, (short)0, acc1, false, false);
    }
    for (int k0 = 0; k0 < 128; k0 += 4) {            // K = 128..255 from ne (row-broadcast)
        v2f bf; bf.x = w1p[128 + k0]; bf.y = w1p[128 + k0 + 1];
        v2f a;  a.x = ne_s[k0 + kb];  a.y = ne_s[k0 + kb + 1];
        acc0 = __builtin_amdgcn_wmma_f32_16x16x4_f32(false, a, false, bf, (short)0, acc0, false, false);
        acc1 = __builtin_amdgcn_wmma_f32_16x16x4_f32(false, a, false, bf, (short)0, acc1, false, false);
    }
    const float bias1 = b1[cc];
#pragma unroll
    for (int v = 0; v < 8; ++v) {                    // D layout: rows v + 8*half
        hbuf[v + 8 * half][cc]      = fmaxf(acc0[v] + bias1, 0.f);
        hbuf[16 + v + 8 * half][cc] = fmaxf(acc1[v] + bias1, 0.f);
    }
    __syncthreads();

    // ---- GEMM2: out = h (32x128) @ W2^T (128x128) + b2 + pv (residual)
    v8f acc2 = {};
    v8f acc3 = {};
    const float* w2p = W2 + (size_t)cc * Hd + kb;
    for (int k0 = 0; k0 < 128; k0 += 4) {
        v2f bf; bf.x = w2p[k0]; bf.y = w2p[k0 + 1];
        v2f a0; a0.x = hbuf[l16][k0 + kb];      a0.y = hbuf[l16][k0 + kb + 1];
        v2f a1; a1.x = hbuf[16 + l16][k0 + kb]; a1.y = hbuf[16 + l16][k0 + kb + 1];
        acc2 = __builtin_amdgcn_wmma_f32_16x16x4_f32(false, a0, false, bf, (short)0, acc2, false, false);
        acc3 = __builtin_amdgcn_wmma_f32_16x16x4_f32(false, a1, false, bf, (short)0, acc3, false, false);
    }
    const float bias2 = b2[cc];
    float* outp = nv + (size_t)node * NV_STRIDE + bofs;
#pragma unroll
    for (int v = 0; v < 8; ++v) {
        const int r0 = v + 8 * half;
        const int r1 = 16 + v + 8 * half;
        outp[r0 * Hd + cc] = acc2[v] + bias2 + pv[r0][cc];
        outp[r1 * Hd + cc] = acc3[v] + bias2 + pv[r1][cc];
    }
}

// out[b][n] = dot(nv[n+1][b][:], Wout[n][:]) + bout[n]
__global__ __launch_bounds__(256) void dag_out_kernel(const float* __restrict__ nv,
                                                      const float* __restrict__ Wout,
                                                      const float* __restrict__ bout,
                                                      float* __restrict__ out) {
    __shared__ float ws[Hd];
    const int n = blockIdx.x;           // 0..20000
    const int t = threadIdx.x;
    if (t < Hd) ws[t] = Wout[(size_t)n * Hd + t];
    __syncthreads();

    const int b   = t >> 2;             // 64 batch rows, 4 lanes each
    const int seg = (t & 3) * 32;
    const float* row = nv + (size_t)(n + 1) * NV_STRIDE + b * Hd + seg;
    float s = 0.f;
#pragma unroll
    for (int h = 0; h < 32; ++h) s += row[h] * ws[seg + h];
    s += __shfl_xor(s, 1, 32);          // wave32 cross-lane reduce (4 lanes per b)
    s += __shfl_xor(s, 2, 32);
    if ((t & 3) == 0) out[(size_t)b * Nnodes + n] = s + bout[n];
}

extern "C" void kernel_launch(void* const* d_in, const int* in_sizes, int n_in,
                              void* d_out, int out_size, void* d_ws, size_t ws_size,
                              hipStream_t stream) {
    (void)in_sizes; (void)n_in; (void)out_size; (void)ws_size;
    const float* embedding      = (const float*)d_in[0];
    const int*   node_indices   = (const int*)d_in[1];
    const int*   parent_indices = (const int*)d_in[2];
    const float* emb_table      = (const float*)d_in[3];
    const float* W1             = (const float*)d_in[4];
    const float* b1             = (const float*)d_in[5];
    const float* W2             = (const float*)d_in[6];
    const float* b2             = (const float*)d_in[7];
    const float* Wout           = (const float*)d_in[8];
    const float* bout           = (const float*)d_in[9];
    float* out = (float*)d_out;
    float* nv  = (float*)d_ws;          // (20002, 64, 128) f32 node state

    dag_init_kernel<<<(Bsz * Hd + 255) / 256, 256, 0, stream>>>(embedding, nv);
    for (int d = 0; d < Dd; ++d)
        dag_depth_kernel<<<dim3(Pn, 2), 256, 0, stream>>>(node_indices, parent_indices,
                                                          emb_table, W1, b1, W2, b2, nv, d);
    dag_out_kernel<<<Nnodes, 256, 0, stream>>>(nv, Wout, bout, out);
}